// PolyConv_53944789238192
// MI455X (gfx1250) — compile-verified
//
#include <hip/hip_runtime.h>

typedef __attribute__((ext_vector_type(2))) float v2f;
typedef __attribute__((ext_vector_type(8))) float v8f;

#define N_ITEMS   1000000
#define DIM       64
#define OUTD      64
#define KTOT      192            // 3*DIM concatenated
#define LDS_STRIDE 196           // padded row stride in dwords (196 % 64 == 4 -> conflict free)
#define W_DWORDS  (OUTD * LDS_STRIDE)            // 12544 dwords
#define XWAVE_DWORDS (16 * LDS_STRIDE)           // 3136 dwords per wave
#define NWAVES    8
#define LDS_DWORDS (W_DWORDS + NWAVES * XWAVE_DWORDS)  // 37632 dwords = 150528 B

// ---- CDNA5 async global->LDS staging (ASYNCcnt path), guarded so the ----
// ---- proven load+store staging remains the fallback on either pass.   ----
#if __has_builtin(__builtin_amdgcn_global_load_async_to_lds_b64)
#define USE_ASYNC_LDS 1
// Builtin parameter type (from compiler diagnostic): generic 'int __vector(2) *'
typedef int v2i_async __attribute__((vector_size(8)));
#define A2I(p) ((v2i_async*)(void*)(p))

__device__ __forceinline__ void wait_async_zero() {
#if __has_builtin(__builtin_amdgcn_s_wait_asynccnt)
    __builtin_amdgcn_s_wait_asynccnt(0);
#else
    asm volatile("s_wait_asynccnt 0x0" ::: "memory");
#endif
}
#endif

__global__ __launch_bounds__(256)
void poly_fused_wmma_kernel(const float* __restrict__ x,
                            const int*   __restrict__ lidx,
                            const int*   __restrict__ ridx,
                            const float* __restrict__ W,
                            const float* __restrict__ bias,
                            float*       __restrict__ out)
{
    extern __shared__ float lds[];
    const int tid  = threadIdx.x;
    const int lane = tid & 31;
    const int wave = tid >> 5;
    const int nsub = lane & 15;     // column/row-within-tile index
    const int half = lane >> 4;     // K-half selector for A/B fragments

    float* xl = lds + W_DWORDS + wave * XWAVE_DWORDS;
    const long rb = (long)blockIdx.x * 128 + (long)wave * 16;
    const int col = lane * 2;       // 8B per lane per source -> 256B coalesced per row

    // ---------------- Stage W (64 x 192) into LDS, padded rows ----------------
#if defined(USE_ASYNC_LDS)
    for (int c = tid; c < (OUTD * KTOT / 2); c += 256) {
        int row = c / (KTOT / 2);
        int wcol = (c % (KTOT / 2)) * 2;
        __builtin_amdgcn_global_load_async_to_lds_b64(
            A2I(W + (long)row * KTOT + wcol),
            A2I(lds + row * LDS_STRIDE + wcol), 0, 0);
    }
#else
    for (int c = tid; c < (OUTD * KTOT / 2); c += 256) {
        int row = c / (KTOT / 2);
        int wcol = (c % (KTOT / 2)) * 2;
        v2f v = *(const v2f*)(W + (long)row * KTOT + wcol);
        *(v2f*)(lds + row * LDS_STRIDE + wcol) = v;
    }
#endif

    // ---------------- Stage this wave's 16 concatenated rows ----------------
    // Row layout in LDS: [ x[n] (64) | x[l] (64) | x[r] (64) ] padded to 196.
    for (int r = 0; r < 16; ++r) {
        long n = rb + r;
        if (n >= N_ITEMS) n = N_ITEMS - 1;   // clamp tail (stores are guarded later)
        long sl = (long)lidx[n];             // uniform -> scalar loads
        long sr = (long)ridx[n];
        float* dst = xl + r * LDS_STRIDE;
#if defined(USE_ASYNC_LDS)
        __builtin_amdgcn_global_load_async_to_lds_b64(A2I(x + n  * DIM + col), A2I(dst + col),            0, 0);
        __builtin_amdgcn_global_load_async_to_lds_b64(A2I(x + sl * DIM + col), A2I(dst + DIM + col),      0, 0);
        __builtin_amdgcn_global_load_async_to_lds_b64(A2I(x + sr * DIM + col), A2I(dst + 2 * DIM + col),  0, 0);
#else
        v2f vx = *(const v2f*)(x + n  * DIM + col);
        v2f vl = *(const v2f*)(x + sl * DIM + col);
        v2f vr = *(const v2f*)(x + sr * DIM + col);
        *(v2f*)(dst + col)            = vx;
        *(v2f*)(dst + DIM + col)      = vl;
        *(v2f*)(dst + 2 * DIM + col)  = vr;
#endif
    }

    // ---------------- Bias-initialized accumulators (C/D layout) ----------------
    // VGPR i, lanes 0-15: (M=i, N=nsub); lanes 16-31: (M=i+8, N=nsub). Col = tile*16+nsub.
    float b0 = bias[ 0 + nsub];
    float b1 = bias[16 + nsub];
    float b2 = bias[32 + nsub];
    float b3 = bias[48 + nsub];
    v8f c0 = {b0,b0,b0,b0,b0,b0,b0,b0};
    v8f c1 = {b1,b1,b1,b1,b1,b1,b1,b1};
    v8f c2 = {b2,b2,b2,b2,b2,b2,b2,b2};
    v8f c3 = {b3,b3,b3,b3,b3,b3,b3,b3};

#if defined(USE_ASYNC_LDS)
    wait_async_zero();               // all async LDS writes landed
#endif
    __syncthreads();

    // ---------------- Main loop: 48 k-steps of K=4 over the 192-dim row -------
    // A frag (16x4):  lane(half,m=nsub): { x_cat[m, 4k+2*half], x_cat[m, 4k+2*half+1] }
    // B frag (4x16):  lane(half,n=nsub): { W[o*16+n, 4k+2*half], W[o*16+n, 4k+2*half+1] }
    #pragma unroll 8
    for (int kk = 0; kk < 48; ++kk) {
        const int kc = kk * 4 + half * 2;
        v2f a  = *(const v2f*)(xl  + nsub * LDS_STRIDE + kc);
        v2f w0 = *(const v2f*)(lds + ( 0 + nsub) * LDS_STRIDE + kc);
        v2f w1 = *(const v2f*)(lds + (16 + nsub) * LDS_STRIDE + kc);
        v2f w2 = *(const v2f*)(lds + (32 + nsub) * LDS_STRIDE + kc);
        v2f w3 = *(const v2f*)(lds + (48 + nsub) * LDS_STRIDE + kc);
        c0 = __builtin_amdgcn_wmma_f32_16x16x4_f32(false, a, false, w0, (short)0, c0, false, false);
        c1 = __builtin_amdgcn_wmma_f32_16x16x4_f32(false, a, false, w1, (short)0, c1, false, false);
        c2 = __builtin_amdgcn_wmma_f32_16x16x4_f32(false, a, false, w2, (short)0, c2, false, false);
        c3 = __builtin_amdgcn_wmma_f32_16x16x4_f32(false, a, false, w3, (short)0, c3, false, false);
    }

    // ---------------- Store 16 x 64 output tile ----------------
    if (rb < N_ITEMS) {              // uniform per wave; N is a multiple of 16 so no partial slabs
        #pragma unroll
        for (int i = 0; i < 8; ++i) {
            float* p = out + (rb + i + half * 8) * OUTD + nsub;
            p[ 0] = c0[i];
            p[16] = c1[i];
            p[32] = c2[i];
            p[48] = c3[i];
        }
    }
}

extern "C" void kernel_launch(void* const* d_in, const int* in_sizes, int n_in,
                              void* d_out, int out_size, void* d_ws, size_t ws_size,
                              hipStream_t stream) {
    const float* x  = (const float*)d_in[0];
    const int*   li = (const int*)d_in[1];
    const int*   ri = (const int*)d_in[2];
    const float* W  = (const float*)d_in[3];
    const float* b  = (const float*)d_in[4];
    float* out = (float*)d_out;

    const int nblocks = (N_ITEMS + 127) / 128;      // 7813
    const size_t shmem = (size_t)LDS_DWORDS * sizeof(float);  // ~147 KB of the 320 KB WGP LDS
    poly_fused_wmma_kernel<<<nblocks, 256, shmem, stream>>>(x, li, ri, W, b, out);
}